// IWF_12532714570025
// MI455X (gfx1250) — compile-verified
//
#include <hip/hip_runtime.h>
#include <hip/hip_bf16.h>
#include <math.h>

typedef float v2f __attribute__((ext_vector_type(2)));
typedef float v8f __attribute__((ext_vector_type(8)));

#define KNN_K 5
#define CHPTS 256               // points per async-staged chunk (4 KB)

// ---------------------------------------------------------------------------
// K1: prep.  pts4[i] = (x,y,z,0)   arrA[i] = (-2x,-2y,-2z, ||p||^2)
// ---------------------------------------------------------------------------
__global__ __launch_bounds__(256) void prep_kernel(const float* __restrict__ pts,
                                                   float4* __restrict__ pts4,
                                                   float4* __restrict__ arrA,
                                                   int N) {
    int i = blockIdx.x * 256 + threadIdx.x;
    if (i < N) {
        float x = pts[3 * i + 0];
        float y = pts[3 * i + 1];
        float z = pts[3 * i + 2];
        float sq = x * x + y * y + z * z;
        pts4[i] = make_float4(x, y, z, 0.0f);
        arrA[i] = make_float4(-2.0f * x, -2.0f * y, -2.0f * z, sq);
    }
}

// Sorted ascending insert into (k0<=..<=k4) held as unsigned bit patterns.
// All values are non-negative finite floats, so unsigned-int ordering ==
// float ordering and v_min_u32/v_max_u32 need no NaN canonicalization.
// No-op for lanes whose c >= k4 (safe under a wave-uniform guard).
__device__ __forceinline__ void ins5u(unsigned c, unsigned& k0, unsigned& k1,
                                      unsigned& k2, unsigned& k3, unsigned& k4) {
    unsigned t = c, n;
    n = min(k0, t); t = max(k0, t); k0 = n;
    n = min(k1, t); t = max(k1, t); k1 = n;
    n = min(k2, t); t = max(k2, t); k2 = n;
    n = min(k3, t); t = max(k3, t); k3 = n;
    k4 = min(k4, t);
}

// ---------------------------------------------------------------------------
// K2: fused pairwise-d2 (WMMA f32 16x16x4) + per-row top-5, with the
// j-point operand stream double-buffered through LDS via async copies.
//   A (16x4) : points j rows = (-2x,-2y,-2z,sq_j)     (from LDS, pipelined)
//   B (4x16) : points i cols = ( x,  y,  z, 1  )      (registers, fixed)
//   C        : sq_i broadcast  =>  D[m][n] = d2 exactly, zero fixup VALU
// C/D layout: lane L -> column n=L%16; VGPR v -> m = v + (L<16 ? 0 : 8).
// ---------------------------------------------------------------------------
__global__ __launch_bounds__(256) void knn_kernel(const float4* __restrict__ pts4,
                                                  const float4* __restrict__ arrA,
                                                  float* __restrict__ knn,
                                                  int N) {
    __shared__ float4 abuf[2][CHPTS];          // 8 KB double buffer

    const int tid = threadIdx.x;
    const int lane = tid & 31;
    const int wave = tid >> 5;
    const int rowBase = blockIdx.x * 128 + wave * 16;
    const int n = lane & 15;
    const bool hi = lane >= 16;
    const int p = rowBase + n;

    // B operand (fixed per wave): VGPR v, lane-half h -> K = v + 2h
    float4 pp = pts4[p];
    v2f b;
    b.x = hi ? pp.z : pp.x;
    b.y = hi ? 1.0f : pp.y;

    const float sqn = arrA[p].w;               // C init: sq of this query point

    unsigned k0 = 0x7149f2cau, k1 = 0x7149f2cau, k2 = 0x7149f2cau,
             k3 = 0x7149f2cau, k4 = 0x7149f2cau;          // bits of 1e30f

    const int nchunks = N / CHPTS;

    // ---- preload chunk 0 into buffer 0 (block-cooperative async copy) ----
    {
        const float4* gp = arrA + tid;                     // chunk 0, 16B/thread
        unsigned ldsaddr = (unsigned)(uintptr_t)&abuf[0][tid];
        asm volatile("global_load_async_to_lds_b128 %0, %1, off"
                     :: "v"(ldsaddr), "v"(gp) : "memory");
        asm volatile("s_wait_asynccnt 0x0" ::: "memory");
        __syncthreads();
    }

    for (int c = 0; c < nchunks; ++c) {
        const int cur = c & 1;

        // overlap: stream next chunk while computing on current one
        if (c + 1 < nchunks) {
            const float4* gp = arrA + (c + 1) * CHPTS + tid;
            unsigned ldsaddr = (unsigned)(uintptr_t)&abuf[cur ^ 1][tid];
            asm volatile("global_load_async_to_lds_b128 %0, %1, off"
                         :: "v"(ldsaddr), "v"(gp) : "memory");
        }

        const float* abf = (const float*)&abuf[cur][0];
        const int lsel = (hi ? 2 : 0);

        // one-deep register pipeline on the A operand (hide ds_load latency)
        v2f a;
        a.x = abf[4 * n + lsel + 0];
        a.y = abf[4 * n + lsel + 1];

#pragma unroll 4
        for (int t = 0; t < CHPTS / 16; ++t) {
            v2f a_cur = a;
            const int tn = (t + 1) & (CHPTS / 16 - 1);     // wraps, branch-free
            const float* apn = abf + 4 * (tn * 16 + n) + lsel;
            a.x = apn[0];
            a.y = apn[1];

            v8f cc = {sqn, sqn, sqn, sqn, sqn, sqn, sqn, sqn};
            cc = __builtin_amdgcn_wmma_f32_16x16x4_f32(
                    /*neg_a=*/false, a_cur, /*neg_b=*/false, b,
                    /*c_mod=*/(short)0, cc, /*reuse_a=*/false, /*reuse_b=*/false);

#pragma unroll
            for (int v = 0; v < 8; ++v) {
                float d2 = cc[v];
                // reference: dist = d2>0 ? sqrt(d2) : 100 -> sentinel 1e4 in d2 space
                float candf = d2 > 0.0f ? d2 : 1.0e4f;
                unsigned cand = __float_as_uint(candf);
                // wave-uniform skip: insert only if some lane improves its top-5
                if (__any(cand < k4))
                    ins5u(cand, k0, k1, k2, k3, k4);
            }
        }

        // next chunk must be resident before buffers swap
        asm volatile("s_wait_asynccnt 0x0" ::: "memory");
        __syncthreads();
    }

    // merge lane L with lane L^16 (same query row, disjoint j halves)
    {
        unsigned p0 = (unsigned)__shfl_xor((int)k0, 16, 32);
        unsigned p1 = (unsigned)__shfl_xor((int)k1, 16, 32);
        unsigned p2 = (unsigned)__shfl_xor((int)k2, 16, 32);
        unsigned p3 = (unsigned)__shfl_xor((int)k3, 16, 32);
        unsigned p4 = (unsigned)__shfl_xor((int)k4, 16, 32);
        ins5u(p0, k0, k1, k2, k3, k4);
        ins5u(p1, k0, k1, k2, k3, k4);
        ins5u(p2, k0, k1, k2, k3, k4);
        ins5u(p3, k0, k1, k2, k3, k4);
        ins5u(p4, k0, k1, k2, k3, k4);
    }

    if (!hi) {
        float* kr = knn + (size_t)p * KNN_K;
        kr[0] = sqrtf(__uint_as_float(k0));
        kr[1] = sqrtf(__uint_as_float(k1));
        kr[2] = sqrtf(__uint_as_float(k2));
        kr[3] = sqrtf(__uint_as_float(k3));
        kr[4] = sqrtf(__uint_as_float(k4));
    }
}

// ---------------------------------------------------------------------------
// K3: finalize (single block, deterministic LDS tree reductions).
// ---------------------------------------------------------------------------
__global__ __launch_bounds__(256) void finalize_kernel(const float* __restrict__ pts,
                                                       const float* __restrict__ knn,
                                                       float* __restrict__ maskArr,
                                                       float* __restrict__ out,
                                                       int N) {
    __shared__ float red[5 * 256];
    __shared__ float scal[8];
    const int tid = threadIdx.x;

    // ---- Phase A: column-min of knn, thr ----
    float cmin[5] = {1e30f, 1e30f, 1e30f, 1e30f, 1e30f};
    for (int i = tid; i < N; i += 256) {
        const float* kr = knn + (size_t)i * KNN_K;
#pragma unroll
        for (int k = 0; k < 5; ++k) cmin[k] = fminf(cmin[k], kr[k]);
    }
#pragma unroll
    for (int k = 0; k < 5; ++k) red[k * 256 + tid] = cmin[k];
    __syncthreads();
    for (int s = 128; s > 0; s >>= 1) {
        if (tid < s) {
#pragma unroll
            for (int k = 0; k < 5; ++k)
                red[k * 256 + tid] = fminf(red[k * 256 + tid], red[k * 256 + tid + s]);
        }
        __syncthreads();
    }
    if (tid == 0)
        scal[0] = (red[0] + red[256] + red[512] + red[768] + red[1024]) * 0.2f;
    __syncthreads();
    const float thr2 = 2.0f * scal[0];

    // ---- Phase B: mask + masked sums -> centroid ----
    float sx = 0.0f, sy = 0.0f, sz = 0.0f, sc = 0.0f;
    for (int i = tid; i < N; i += 256) {
        const float* kr = knn + (size_t)i * KNN_K;
        int cnt = 0;
#pragma unroll
        for (int k = 0; k < 5; ++k) cnt += (kr[k] > thr2) ? 1 : 0;
        float mf = (cnt <= 3) ? 1.0f : 0.0f;   // cnt/5 < 0.75
        maskArr[i] = mf;
        sx += pts[3 * i + 0] * mf;
        sy += pts[3 * i + 1] * mf;
        sz += pts[3 * i + 2] * mf;
        sc += mf;
    }
    red[0 * 256 + tid] = sx;
    red[1 * 256 + tid] = sy;
    red[2 * 256 + tid] = sz;
    red[3 * 256 + tid] = sc;
    __syncthreads();
    for (int s = 128; s > 0; s >>= 1) {
        if (tid < s) {
#pragma unroll
            for (int k = 0; k < 4; ++k)
                red[k * 256 + tid] += red[k * 256 + tid + s];
        }
        __syncthreads();
    }
    if (tid == 0) {
        float inv = 1.0f / red[3 * 256];
        scal[1] = red[0] * inv;
        scal[2] = red[256] * inv;
        scal[3] = red[512] * inv;
    }
    __syncthreads();
    const float cx = scal[1], cy = scal[2], cz = scal[3];

    // ---- Phase C: max masked norm ----
    float mx = 0.0f;
    for (int i = tid; i < N; i += 256) {
        float dx = pts[3 * i + 0] - cx;
        float dy = pts[3 * i + 1] - cy;
        float dz = pts[3 * i + 2] - cz;
        float nrm = sqrtf(dx * dx + dy * dy + dz * dz);
        mx = fmaxf(mx, nrm * maskArr[i]);
    }
    red[tid] = mx;
    __syncthreads();
    for (int s = 128; s > 0; s >>= 1) {
        if (tid < s) red[tid] = fmaxf(red[tid], red[tid + s]);
        __syncthreads();
    }
    if (tid == 0) scal[4] = red[0];
    __syncthreads();
    const float invm = 1.0f / scal[4];

    // ---- Phase D: write output ----
    for (int i = tid; i < N; i += 256) {
        float mf = maskArr[i];
        out[3 * i + 0] = (pts[3 * i + 0] - cx) * invm * mf;
        out[3 * i + 1] = (pts[3 * i + 1] - cy) * invm * mf;
        out[3 * i + 2] = (pts[3 * i + 2] - cz) * invm * mf;
    }
}

// ---------------------------------------------------------------------------
extern "C" void kernel_launch(void* const* d_in, const int* in_sizes, int n_in,
                              void* d_out, int out_size, void* d_ws, size_t ws_size,
                              hipStream_t stream) {
    const float* pts = (const float*)d_in[0];
    const int N = in_sizes[0] / 3;          // 16384

    // workspace layout (floats)
    float* wsf = (float*)d_ws;
    float4* pts4 = (float4*)(wsf);                       // N*4
    float4* arrA = (float4*)(wsf + (size_t)N * 4);       // N*4
    float* knn   = wsf + (size_t)N * 8;                  // N*5
    float* maskA = wsf + (size_t)N * 13;                 // N
    float* out   = (float*)d_out;

    prep_kernel<<<(N + 255) / 256, 256, 0, stream>>>(pts, pts4, arrA, N);
    knn_kernel<<<N / 128, 256, 0, stream>>>(pts4, arrA, knn, N);
    finalize_kernel<<<1, 256, 0, stream>>>(pts, knn, maskA, out, N);
}